// SSG_2319282340203
// MI455X (gfx1250) — compile-verified
//
#include <hip/hip_runtime.h>
#include <hip/hip_bf16.h>
#include <math.h>

// Problem constants (match reference setup_inputs)
#define BB   4
#define NN   8192
#define CC   64
#define KK   8
#define INV_SCALE 0.35355339059327373f   // 1/sqrt(8)

typedef float v2f __attribute__((ext_vector_type(2)));
typedef float v8f __attribute__((ext_vector_type(8)));

// ---------------------------------------------------------------------------
// 32-lane shuffle helpers (wave32 on gfx1250)
// ---------------------------------------------------------------------------
__device__ inline unsigned long long shfl_xor_u64(unsigned long long v, int mask) {
    unsigned lo = (unsigned)v;
    unsigned hi = (unsigned)(v >> 32);
    lo = (unsigned)__shfl_xor((int)lo, mask, 32);
    hi = (unsigned)__shfl_xor((int)hi, mask, 32);
    return ((unsigned long long)hi << 32) | lo;
}

// Order-preserving float -> uint mapping (handles negative keys).
__device__ inline unsigned float_to_ord(float f) {
    unsigned ub = __float_as_uint(f);
    return ub ^ ((ub & 0x80000000u) ? 0xFFFFFFFFu : 0x80000000u);
}

// ---------------------------------------------------------------------------
// Kernel 1: WMMA-based brute-force KNN.
// key[M][N] = |c_M|^2 - 2 q_N . c_M   (orders identically to d2 per query)
// computed as one V_WMMA_F32_16X16X4_F32 per 16-candidate x 16-query tile:
//   A (16x4) = [cx, cy, cz, |c|^2]   B (4x16) = [-2qx, -2qy, -2qz, 1]^T
// Wave w owns 16 queries; D layout puts 8 candidates of a lane's own query
// in that lane's 8 acc slots -> per-lane top-8 via a branchless select
// network, then a pairwise (lane n <-> n+16) 8-round pop merge.
// Block = 8 waves = 128 queries, sharing candidate tiles staged in LDS.
// ---------------------------------------------------------------------------
__global__ __launch_bounds__(256) void knn_wmma_kernel(
    const float* __restrict__ coords,   // (B, N, 3)
    int* __restrict__ knn)              // (B, N, 8)
{
    __shared__ float4 sC[256];          // candidate (x, y, z, |c|^2)

    const int blocks_per_batch = NN / 128;
    const int b     = blockIdx.x / blocks_per_batch;
    const int q0    = (blockIdx.x % blocks_per_batch) * 128;
    const int tid   = threadIdx.x;
    const int lane  = tid & 31;
    const int wave  = tid >> 5;
    const int n     = lane & 15;        // this lane's query (within wave)
    const int hi    = lane >> 4;        // candidate half select

    const float* Cd = coords + (size_t)b * NN * 3;

    // Loop-invariant B fragment for this lane's query.
    const int q = q0 + wave * 16 + n;
    const float qx = Cd[(size_t)q * 3 + 0];
    const float qy = Cd[(size_t)q * 3 + 1];
    const float qz = Cd[(size_t)q * 3 + 2];
    v2f bfrag;
    if (hi == 0) { bfrag.x = -2.0f * qx; bfrag.y = -2.0f * qy; }
    else         { bfrag.x = -2.0f * qz; bfrag.y = 1.0f;       }

    // Per-lane sorted top-8 of key (ascending). Candidate indices are unique
    // per lane: lane (n,hi) only sees j with (j%16) in [8*hi, 8*hi+8).
    float bd[8];
    int   bi[8];
#pragma unroll
    for (int s = 0; s < 8; ++s) { bd[s] = 3.4e38f; bi[s] = -1; }

    const v8f zero = (v8f){0.f, 0.f, 0.f, 0.f, 0.f, 0.f, 0.f, 0.f};

    for (int base = 0; base < NN; base += 256) {
        __syncthreads();
        {
            const int j = base + tid;
            const float x = Cd[(size_t)j * 3 + 0];
            const float y = Cd[(size_t)j * 3 + 1];
            const float z = Cd[(size_t)j * 3 + 2];
            sC[tid] = make_float4(x, y, z, fmaf(x, x, fmaf(y, y, z * z)));
        }
        __syncthreads();
        if (base + 256 < NN)  // prefetch next coordinate tile into near caches
            __builtin_prefetch(&Cd[(size_t)(base + 256 + tid) * 3], 0, 3);

#pragma unroll 2
        for (int sub = 0; sub < 16; ++sub) {
            // A fragment: lane picks candidate (sub*16 + n);
            // hi=0 -> (cx,cy), hi=1 -> (cz,|c|^2)
            const float2* cp = (const float2*)&sC[sub * 16 + n] + hi;
            v2f a; a.x = cp->x; a.y = cp->y;

            v8f keyv = __builtin_amdgcn_wmma_f32_16x16x4_f32(
                false, a, false, bfrag, (short)0, zero, false, false);

            // Level-1 screen: min of the 8 fresh keys vs current threshold.
            const float mn = fminf(
                fminf(fminf(keyv[0], keyv[1]), fminf(keyv[2], keyv[3])),
                fminf(fminf(keyv[4], keyv[5]), fminf(keyv[6], keyv[7])));
            if (mn < bd[7]) {
                const int cbase = base + sub * 16 + 8 * hi;
#pragma unroll
                for (int v = 0; v < 8; ++v) {
                    const float kd = keyv[v];
                    if (kd < bd[7]) {
                        const int j = cbase + v;
                        // Branchless sorted insert (select network, no
                        // data-dependent control flow inside).
                        bool clt[7];
#pragma unroll
                        for (int s = 0; s < 7; ++s) clt[s] = kd < bd[s];
                        bd[7] = clt[6] ? bd[6] : kd;
                        bi[7] = clt[6] ? bi[6] : j;
#pragma unroll
                        for (int s = 6; s >= 1; --s) {
                            const bool cu = clt[s - 1];
                            const bool ch = clt[s];
                            bd[s] = cu ? bd[s - 1] : (ch ? kd : bd[s]);
                            bi[s] = cu ? bi[s - 1] : (ch ? j  : bi[s]);
                        }
                        bd[0] = clt[0] ? kd : bd[0];
                        bi[0] = clt[0] ? j  : bi[0];
                    }
                }
            }
        }
    }

    // Pairwise merge: lanes n and n+16 hold disjoint candidate sets for the
    // same query. Pop the 8 global minima.
    int res[8];
#pragma unroll
    for (int rnd = 0; rnd < 8; ++rnd) {
        unsigned long long key =
            ((unsigned long long)float_to_ord(bd[0]) << 32) | (unsigned)bi[0];
        const unsigned long long other = shfl_xor_u64(key, 16);
        const unsigned long long win = (other < key) ? other : key;
        const bool pop = (win == key);
#pragma unroll
        for (int s = 0; s < 7; ++s) {
            bd[s] = pop ? bd[s + 1] : bd[s];
            bi[s] = pop ? bi[s + 1] : bi[s];
        }
        if (pop) { bd[7] = 3.4e38f; bi[7] = -1; }
        res[rnd] = (int)(unsigned)win;
    }
    if (hi == 0) {
        int* op = knn + ((size_t)b * NN + q) * KK;
        ((int4*)op)[0] = make_int4(res[0], res[1], res[2], res[3]);
        ((int4*)op)[1] = make_int4(res[4], res[5], res[6], res[7]);
    }
}

// ---------------------------------------------------------------------------
// Kernel 2: gather + dual WMMA GEMM + softmax(K) + weighted sum + residual+LN
// One 256-thread block (8 waves) per tile of 16 points (= 128 neighbor rows).
// Wave w owns rows 16w..16w+15; 4 column tiles of 16 channels each.
// D-layout of wmma_f32_16x16x4: lane's 8 acc slots == the 8 K-neighbors of a
// fixed (point, channel) -> softmax over K is purely in-register per lane.
// ---------------------------------------------------------------------------
__global__ __launch_bounds__(256) void attend_ln_kernel(
    const float* __restrict__ features,  // (B, N, C)
    const float* __restrict__ coords,    // (B, N, 3)
    const float* __restrict__ W_ft,      // (C, C)
    const float* __restrict__ b_ft,      // (C,)
    const float* __restrict__ W_coord,   // (3, C)
    const float* __restrict__ b_coord,   // (C,)
    const float* __restrict__ W_feat,    // (C, C)
    const float* __restrict__ b_feat,    // (C,)
    const float* __restrict__ ln_g,      // (C,)
    const float* __restrict__ ln_b,      // (C,)
    const int*   __restrict__ knn,       // (B, N, 8)
    float* __restrict__ out)             // (B, N, C)
{
    // Padded strides to dodge LDS bank conflicts on column reads.
    __shared__ float sNbr[128][68];      // gathered neighbor features
    __shared__ float sFeat[16][68];      // query-point features
    __shared__ float sWft[64][65];       // W_ft
    __shared__ float sWfe[64][65];       // W_feat
    __shared__ float sRelC[128][4];      // relative coords (xyz, pad)
    __shared__ float sWc[3][64];
    __shared__ float sBft[64], sBc[64], sBfe[64], sG[64], sBeta[64];

    const int tiles_per_batch = NN / 16;
    const int b   = blockIdx.x / tiles_per_batch;
    const int p0  = (blockIdx.x % tiles_per_batch) * 16;
    const int tid = threadIdx.x;
    const int lane = tid & 31;
    const int wave = tid >> 5;
    const int m  = lane & 15;   // fragment row/col within 16
    const int hi = lane >> 4;   // half-wave select

    const float* F  = features + (size_t)b * NN * CC;
    const float* Cd = coords   + (size_t)b * NN * 3;
    const int*   I  = knn + ((size_t)b * NN + p0) * KK;   // 128 indices

    // ---- stage weights & vectors ----
    for (int i = tid; i < 64 * 64; i += 256) {
        sWft[i >> 6][i & 63] = W_ft[i];
        sWfe[i >> 6][i & 63] = W_feat[i];
    }
    if (tid < 3 * 64) sWc[tid / 64][tid & 63] = W_coord[tid];
    if (tid < 64) {
        sBft[tid]  = b_ft[tid];
        sBc[tid]   = b_coord[tid];
        sBfe[tid]  = b_feat[tid];
        sG[tid]    = ln_g[tid];
        sBeta[tid] = ln_b[tid];
    }
    for (int i = tid; i < 16 * CC; i += 256) {
        const int pp = i >> 6, c = i & 63;
        sFeat[pp][c] = F[(size_t)(p0 + pp) * CC + c];
    }
    // ---- gather neighbor rows (2 threads per row, 32 ch each, float4) ----
    {
        const int r    = tid >> 1;      // 0..127
        const int half = tid & 1;       // 0/1
        const int src  = I[r];
        const float4* sp = (const float4*)(F + (size_t)src * CC) + half * 8;
        float4* dp = (float4*)(&sNbr[r][0]) + half * 8;
#pragma unroll
        for (int qq = 0; qq < 8; ++qq) dp[qq] = sp[qq];
        if (half == 0) {
            const int pg = p0 + (r >> 3);
            sRelC[r][0] = Cd[(size_t)src * 3 + 0] - Cd[(size_t)pg * 3 + 0];
            sRelC[r][1] = Cd[(size_t)src * 3 + 1] - Cd[(size_t)pg * 3 + 1];
            sRelC[r][2] = Cd[(size_t)src * 3 + 2] - Cd[(size_t)pg * 3 + 2];
        }
    }
    __syncthreads();

    // ---- dual GEMM: acc1 = nbr_f @ W_ft ; acc2 = (nbr_f - feat) @ W_feat ----
    v8f acc1[4], acc2[4];
#pragma unroll
    for (int t = 0; t < 4; ++t) {
        acc1[t] = (v8f){0.f, 0.f, 0.f, 0.f, 0.f, 0.f, 0.f, 0.f};
        acc2[t] = (v8f){0.f, 0.f, 0.f, 0.f, 0.f, 0.f, 0.f, 0.f};
    }
    const int rA = wave * 16 + m;        // A-matrix row this lane feeds
    const int pl = rA >> 3;              // local point of that row
#pragma unroll
    for (int kt = 0; kt < 16; ++kt) {
        const int ka = kt * 4 + hi * 2;  // K offset per A/B layout
        v2f a1, a2;
        a1.x = sNbr[rA][ka];
        a1.y = sNbr[rA][ka + 1];
        a2.x = a1.x - sFeat[pl][ka];
        a2.y = a1.y - sFeat[pl][ka + 1];
#pragma unroll
        for (int t = 0; t < 4; ++t) {
            const int nn = t * 16 + m;
            v2f bf, be;
            bf.x = sWft[ka][nn];  bf.y = sWft[ka + 1][nn];
            be.x = sWfe[ka][nn];  be.y = sWfe[ka + 1][nn];
            acc1[t] = __builtin_amdgcn_wmma_f32_16x16x4_f32(
                false, a1, false, bf, (short)0, acc1[t], false, false);
            acc2[t] = __builtin_amdgcn_wmma_f32_16x16x4_f32(
                false, a2, false, be, (short)0, acc2[t], false, false);
        }
    }

    // ---- epilogue: score, softmax over K (in-register), weighted sum ----
    const int pD = wave * 2 + hi;        // local point this lane's accs belong to
    const int gp = p0 + pD;              // global point
    float o[4];
#pragma unroll
    for (int t = 0; t < 4; ++t) {
        const int c = t * 16 + m;
        const float w0 = sWc[0][c], w1 = sWc[1][c], w2 = sWc[2][c];
        float sc[8];
        float mx = -3.4e38f;
#pragma unroll
        for (int v = 0; v < 8; ++v) {
            const int rr = pD * 8 + v;
            const float scoord = fmaf(sRelC[rr][0], w0,
                                 fmaf(sRelC[rr][1], w1,
                                 fmaf(sRelC[rr][2], w2, sBc[c])));
            const float t2 = acc2[t][v] + sBfe[c];
            const float s  = scoord * t2 * INV_SCALE;
            sc[v] = s;
            mx = fmaxf(mx, s);
        }
        float sum = 0.f;
        float e[8];
#pragma unroll
        for (int v = 0; v < 8; ++v) { e[v] = __expf(sc[v] - mx); sum += e[v]; }
        const float inv = 1.f / sum;
        float acc = 0.f;
#pragma unroll
        for (int v = 0; v < 8; ++v)
            acc = fmaf(e[v] * inv, acc1[t][v] + sBft[c], acc);
        o[t] = acc + sFeat[pD][c];       // residual
    }

    // ---- LayerNorm over 64 channels: 4 regs x 16-lane shuffle tree ----
    float s1 = o[0] + o[1] + o[2] + o[3];
    float s2 = o[0] * o[0] + o[1] * o[1] + o[2] * o[2] + o[3] * o[3];
#pragma unroll
    for (int off = 8; off >= 1; off >>= 1) {
        s1 += __shfl_xor(s1, off, 32);
        s2 += __shfl_xor(s2, off, 32);
    }
    const float mean = s1 * (1.0f / 64.0f);
    const float var  = s2 * (1.0f / 64.0f) - mean * mean;
    const float rstd = rsqrtf(var + 1e-5f);

    float* orow = out + ((size_t)b * NN + gp) * CC;
#pragma unroll
    for (int t = 0; t < 4; ++t) {
        const int c = t * 16 + m;
        orow[c] = (o[t] - mean) * rstd * sG[c] + sBeta[c];
    }
}

// ---------------------------------------------------------------------------
// Launcher
// ---------------------------------------------------------------------------
extern "C" void kernel_launch(void* const* d_in, const int* in_sizes, int n_in,
                              void* d_out, int out_size, void* d_ws, size_t ws_size,
                              hipStream_t stream) {
    const float* features = (const float*)d_in[0];
    const float* coords   = (const float*)d_in[1];
    const float* W_ft     = (const float*)d_in[2];
    const float* b_ft     = (const float*)d_in[3];
    const float* W_coord  = (const float*)d_in[4];
    const float* b_coord  = (const float*)d_in[5];
    const float* W_feat   = (const float*)d_in[6];
    const float* b_feat   = (const float*)d_in[7];
    const float* ln_g     = (const float*)d_in[8];
    const float* ln_b     = (const float*)d_in[9];
    float* out = (float*)d_out;

    int* knn = (int*)d_ws;   // (B, N, 8) int32 = 1 MB scratch

    knn_wmma_kernel<<<BB * (NN / 128), 256, 0, stream>>>(coords, knn);
    attend_ln_kernel<<<BB * (NN / 16), 256, 0, stream>>>(
        features, coords, W_ft, b_ft, W_coord, b_coord,
        W_feat, b_feat, ln_g, ln_b, knn, out);
}